// StackedLSTM_3796751090203
// MI455X (gfx1250) — compile-verified
//
#include <hip/hip_runtime.h>

typedef __attribute__((ext_vector_type(2))) float v2f;
typedef __attribute__((ext_vector_type(8))) float v8f;

#define BB 32      // batch
#define TT 512     // time steps
#define DD 512     // input dim
#define HH 512     // hidden dim
#define MROWS (BB*TT)   // 16384 rows for the big GEMMs
#define LDSW 516   // padded LDS row stride (516 % 64 == 4 -> conflict-free frags)

// ---------------------------------------------------------------------------
// fp32 WMMA: D(16x16) = A(16x4) * B(4x16) + C.
// A fragment (lane l): row = l%16, k = 2*(l/16) + {0,1}  -> one float2 load
// B fragment (lane l): col = l%16, k = 2*(l/16) + {0,1}  -> identical mapping
//   (so row-major MxK "A" and row-major NxK "W" both load the same way)
// C/D fragment (lane l, vgpr v): m = v + 8*(l/16), n = l%16
// ---------------------------------------------------------------------------
__device__ __forceinline__ v8f wmma4(v2f a, v2f b, v8f c) {
  return __builtin_amdgcn_wmma_f32_16x16x4_f32(
      /*neg_a=*/false, a, /*neg_b=*/false, b,
      /*c_mod=*/(short)0, c, /*reuse_a=*/false, /*reuse_b=*/false);
}

__device__ __forceinline__ float sigm(float x) { return 1.0f / (1.0f + __expf(-x)); }
// tanh(x) = 2*sigmoid(2x) - 1: saturates to +/-1 correctly, NaN-propagating,
// avoids the branchy libm tanhf on the sequential critical path.
__device__ __forceinline__ float ftanh(float x) {
  return __builtin_fmaf(2.0f, sigm(2.0f * x), -1.0f);
}

// ---------------------------------------------------------------------------
// Transpose x: (B,T,DI) -> (T,B,DI) so GEMM A-operand rows are contiguous.
// ---------------------------------------------------------------------------
__global__ void __launch_bounds__(256)
transpose_x(const float* __restrict__ x, float* __restrict__ xt)
{
  int r = blockIdx.x;            // r = t*B + b
  int t = r / BB, b = r % BB;
  const float* src = x + ((long)b * TT + t) * DD;
  float* dst = xt + (long)r * DD;
  for (int k = threadIdx.x; k < DD; k += blockDim.x) dst[k] = src[k];
}

// ---------------------------------------------------------------------------
// Big time-parallel GEMM:  OUT[r, n] = sum_k A[r, k] * W[n*ldw + k] + bias(n)
//   A: (MROWS, 512) row-major; W rows are output columns (ref uses W.T)
//   OUT: (MROWS, N) = (T, B, N)
// Each wave computes a 32(M) x 64(N) tile: acc[2][4] of v8f.
// Block = 8 waves => 32 x 512 of N per block. grid = (N/512, MROWS/32).
// ---------------------------------------------------------------------------
__global__ void __launch_bounds__(256)
gemm_pre(const float* __restrict__ A, const float* __restrict__ W,
         const float* __restrict__ bias0, const float* __restrict__ bias1,
         float* __restrict__ OUT, int N, int ldw)
{
  const int wave = threadIdx.x >> 5;
  const int lane = threadIdx.x & 31;
  const int lr = lane & 15;
  const int lh = lane >> 4;
  const int koff = 2 * lh;
  const int nt0 = (blockIdx.x * 8 + wave) * 4;  // first of 4 N-tiles (16 wide)
  const int m0  = blockIdx.y * 32;

  const float* arow[2];
#pragma unroll
  for (int mt = 0; mt < 2; ++mt)
    arow[mt] = A + (long)(m0 + mt * 16 + lr) * DD;

  const float* wrow[4];
#pragma unroll
  for (int nt = 0; nt < 4; ++nt)
    wrow[nt] = W + (long)((nt0 + nt) * 16 + lr) * ldw;

  const v8f vzero = {0.f, 0.f, 0.f, 0.f, 0.f, 0.f, 0.f, 0.f};
  v8f acc[2][4];
#pragma unroll
  for (int mt = 0; mt < 2; ++mt)
#pragma unroll
    for (int nt = 0; nt < 4; ++nt) acc[mt][nt] = vzero;

#pragma unroll 2
  for (int k0 = 0; k0 < DD; k0 += 4) {
    v2f a[2], b[4];
#pragma unroll
    for (int mt = 0; mt < 2; ++mt)
      a[mt] = *(const v2f*)(arow[mt] + k0 + koff);
#pragma unroll
    for (int nt = 0; nt < 4; ++nt)
      b[nt] = *(const v2f*)(wrow[nt] + k0 + koff);
#pragma unroll
    for (int mt = 0; mt < 2; ++mt)
#pragma unroll
      for (int nt = 0; nt < 4; ++nt)
        acc[mt][nt] = wmma4(a[mt], b[nt], acc[mt][nt]);
  }

#pragma unroll
  for (int nt = 0; nt < 4; ++nt) {
    const int n = (nt0 + nt) * 16 + lr;          // D-matrix n = lane%16
    const float bv = bias0[n] + (bias1 ? bias1[n] : 0.f);
#pragma unroll
    for (int mt = 0; mt < 2; ++mt) {
#pragma unroll
      for (int v = 0; v < 8; ++v) {
        const int m = m0 + mt * 16 + lh * 8 + v; // D-matrix m = v + 8*(lane/16)
        OUT[(long)m * N + n] = acc[mt][nt][v] + bv;
      }
    }
  }
}

// ---------------------------------------------------------------------------
// Persistent recurrent kernel (one launch per layer).
//   P:    (T, B, NG*H) precomputed input contribution + bias
//   Wr:   gate weights; row n at Wr + n*ldw + woff, K = H (the h_prev block)
//   hbuf/cbuf: (T+1, B, H); slot 0 pre-zeroed; step t reads slot t, writes t+1
//   hlow/clow (NG==5): lower layer's (T+1,B,H) buffers (slot t+1 = time t)
// 64 waves (8 blocks x 8 waves): wave = (strip j in [0,32)) x (batch half m0).
// h_{t-1} (64 KB) is staged into LDS each step (padded stride -> no bank
// conflicts on the ds_load_b64 A-fragments). Per-timestep sync: monotonic
// global atomic barrier (all 8 blocks trivially co-resident).
// ---------------------------------------------------------------------------
template <int NG>
__global__ void __launch_bounds__(256)
lstm_rec(const float* __restrict__ P, const float* __restrict__ Wr,
         int ldw, int woff,
         const float* __restrict__ hlow, const float* __restrict__ clow,
         float* __restrict__ hbuf, float* __restrict__ cbuf,
         int* __restrict__ bar)
{
  __shared__ float hlds[BB * LDSW];   // ~64.5 KB of the 320 KB WGP LDS

  const int wave = threadIdx.x >> 5;
  const int lane = threadIdx.x & 31;
  const int lr = lane & 15;
  const int lh = lane >> 4;
  const int koff = 2 * lh;
  const int wg = blockIdx.x * 8 + wave;   // 0..63
  const int j  = wg >> 1;                 // hidden strip (16 cols)
  const int m0 = (wg & 1) * 16;           // batch half
  const int n  = j * 16 + lr;             // this lane's hidden column
  const int nblocks = gridDim.x;

  const float* wrow[NG];
#pragma unroll
  for (int g = 0; g < NG; ++g)
    wrow[g] = Wr + (long)(g * HH + j * 16 + lr) * ldw + woff;

  const float* arow = &hlds[(m0 + lr) * LDSW];
  const v8f vzero = {0.f, 0.f, 0.f, 0.f, 0.f, 0.f, 0.f, 0.f};

  for (int t = 0; t < TT; ++t) {
    // ---- stage h_{t-1} (B x H) into LDS, padded row stride ----
    {
      const float4* src4 = (const float4*)(hbuf + (long)t * BB * HH);
      for (int i = threadIdx.x; i < (BB * HH) / 4; i += 256) {
        const int r = i >> 7;            // row (H/4 = 128 float4 per row)
        const int c = (i & 127) << 2;    // column (floats)
        *(float4*)&hlds[r * LDSW + c] = src4[i];
      }
    }
    __syncthreads();

    v8f acc[NG];
#pragma unroll
    for (int g = 0; g < NG; ++g) acc[g] = vzero;

#pragma unroll 4
    for (int k0 = 0; k0 < HH; k0 += 4) {
      const v2f a = *(const v2f*)(arow + k0 + koff);    // ds_load_b64
#pragma unroll
      for (int g = 0; g < NG; ++g) {
        const v2f b = *(const v2f*)(wrow[g] + k0 + koff);
        acc[g] = wmma4(a, b, acc[g]);
      }
    }

    const float* Pt     = P + (long)t * BB * (NG * HH);
    const float* cprow  = cbuf + (long)t * BB * HH;
    float* hout  = hbuf + (long)(t + 1) * BB * HH;
    float* cout_ = cbuf + (long)(t + 1) * BB * HH;

#pragma unroll
    for (int v = 0; v < 8; ++v) {
      const int m = m0 + lh * 8 + v;            // batch row for this element
      const long pbase = (long)m * (NG * HH) + n;
      const float cprev = cprow[(long)m * HH + n];
      float cnew, hnew;
      if constexpr (NG == 4) {
        const float gi = acc[0][v] + Pt[pbase + 0 * HH];
        const float gf = acc[1][v] + Pt[pbase + 1 * HH];
        const float gg = acc[2][v] + Pt[pbase + 2 * HH];
        const float go = acc[3][v] + Pt[pbase + 3 * HH];
        cnew = sigm(gi) * ftanh(gg) + sigm(gf) * cprev;
        hnew = sigm(go) * ftanh(cnew);
      } else {
        const float gi  = acc[0][v] + Pt[pbase + 0 * HH];
        const float gfp = acc[1][v] + Pt[pbase + 1 * HH];
        const float gfl = acc[2][v] + Pt[pbase + 2 * HH];
        const float gu  = acc[3][v] + Pt[pbase + 3 * HH];
        const float go  = acc[4][v] + Pt[pbase + 4 * HH];
        const float cl  = clow[(long)(t + 1) * BB * HH + (long)m * HH + n];
        cnew = cprev * sigm(gfp + 1.f) + cl * sigm(gfl + 1.f) + ftanh(gu) * sigm(gi);
        hnew = sigm(go) * ftanh(cnew);
      }
      hout[(long)m * HH + n] = hnew;
      cout_[(long)m * HH + n] = cnew;
    }

    if (t != TT - 1) {
      __threadfence();            // release our h/c stores
      __syncthreads();            // also protects hlds before next-step copy
      if (threadIdx.x == 0) {
        atomicAdd(bar, 1);
        const int target = (t + 1) * nblocks;
        while (atomicAdd(bar, 0) < target) __builtin_amdgcn_s_sleep(1);
      }
      __syncthreads();
      __threadfence();            // acquire other blocks' stores
    }
  }
}

// ---------------------------------------------------------------------------
// Final writeout: top layer (T+1,B,H) buffers -> d_out:
//   hs (B,T,H) | cs (B,T,H) | h_last (B,H) | c_last (B,H)
// ---------------------------------------------------------------------------
__global__ void __launch_bounds__(256)
writeout(const float* __restrict__ hbuf, const float* __restrict__ cbuf,
         float* __restrict__ out)
{
  const long BTH = (long)BB * TT * HH;
  long idx = (long)blockIdx.x * blockDim.x + threadIdx.x;
  if (idx >= BTH) return;
  const int h = (int)(idx % HH);
  const long r = idx / HH;
  const int t = (int)(r % TT);
  const int b = (int)(r / TT);
  const long src = (long)(t + 1) * BB * HH + (long)b * HH + h;
  const float hv = hbuf[src];
  const float cv = cbuf[src];
  out[idx] = hv;
  out[BTH + idx] = cv;
  if (t == TT - 1) {
    out[2 * BTH + (long)b * HH + h] = hv;
    out[2 * BTH + (long)BB * HH + (long)b * HH + h] = cv;
  }
}

// ---------------------------------------------------------------------------
extern "C" void kernel_launch(void* const* d_in, const int* in_sizes, int n_in,
                              void* d_out, int out_size, void* d_ws, size_t ws_size,
                              hipStream_t stream)
{
  const float* x    = (const float*)d_in[0];
  const float* w_ih = (const float*)d_in[1];
  const float* w_hh = (const float*)d_in[2];
  const float* b_ih = (const float*)d_in[3];
  const float* b_hh = (const float*)d_in[4];
  const float* ca_w = (const float*)d_in[5];
  const float* ca_b = (const float*)d_in[6];
  float* out = (float*)d_out;

  // Workspace layout (floats):
  //   P  : (T,B,5H) | xt : (T,B,DI) | hA,cA,hB,cB : (T+1,B,H) | bar
  float* ws = (float*)d_ws;
  const long Psz  = (long)TT * BB * 5 * HH;
  const long Xsz  = (long)TT * BB * DD;
  const long HBsz = (long)(TT + 1) * BB * HH;
  float* P  = ws;
  float* xt = P + Psz;
  float* hA = xt + Xsz;
  float* cA = hA + HBsz;
  float* hB = cA + HBsz;
  float* cB = hB + HBsz;
  int*   bar = (int*)(cB + HBsz);

  const dim3 blk(256, 1, 1);

  // ---- layer 0 (plain LSTM) ----
  hipMemsetAsync(hA, 0, (size_t)BB * HH * sizeof(float), stream);
  hipMemsetAsync(cA, 0, (size_t)BB * HH * sizeof(float), stream);
  hipMemsetAsync(bar, 0, sizeof(int), stream);
  transpose_x<<<dim3(MROWS, 1, 1), blk, 0, stream>>>(x, xt);
  gemm_pre<<<dim3((4 * HH) / 512, MROWS / 32, 1), blk, 0, stream>>>(
      xt, w_ih, b_ih, b_hh, P, 4 * HH, DD);
  lstm_rec<4><<<dim3(8, 1, 1), blk, 0, stream>>>(
      P, w_hh, HH, 0, nullptr, nullptr, hA, cA, bar);

  // ---- layers 1..3 (CALSTM, fuse='add') ----
  float* hl = hA; float* clo = cA;   // lower-layer buffers
  float* hc = hB; float* cc = cB;    // current-layer buffers
  for (int l = 0; l < 3; ++l) {
    const float* Wl = ca_w + (long)l * (5 * HH) * (2 * HH);
    const float* bl = ca_b + (long)l * (5 * HH);
    hipMemsetAsync(hc, 0, (size_t)BB * HH * sizeof(float), stream);
    hipMemsetAsync(cc, 0, (size_t)BB * HH * sizeof(float), stream);
    hipMemsetAsync(bar, 0, sizeof(int), stream);
    // P = h_low @ Wl[:, :H].T + bl   (h_low rows = slots 1..T of lower buffer)
    gemm_pre<<<dim3((5 * HH) / 512, MROWS / 32, 1), blk, 0, stream>>>(
        hl + (long)BB * HH, Wl, bl, nullptr, P, 5 * HH, 2 * HH);
    // recurrent part uses Wl[:, H:2H] (row stride 2H, column offset H)
    lstm_rec<5><<<dim3(8, 1, 1), blk, 0, stream>>>(
        P, Wl, 2 * HH, HH, hl, clo, hc, cc, bar);
    float* th = hl; hl = hc; hc = th;
    float* tc = clo; clo = cc; cc = tc;
  }

  // ---- writeout ----
  const long BTH = (long)BB * TT * HH;
  writeout<<<dim3((unsigned)((BTH + 255) / 256), 1, 1), blk, 0, stream>>>(hl, clo, out);
}